// Block_10230612099159
// MI455X (gfx1250) — compile-verified
//
#include <hip/hip_runtime.h>
#include <hip/hip_bf16.h>

// CDNA5 (gfx1250) transformer block, fp32 throughout.
// - All matmuls on V_WMMA_F32_16X16X4_F32 (exact fp32 matrix pipe), wave32.
// - GEMM mainloop double-buffered via GLOBAL_LOAD_ASYNC_TO_LDS (ASYNCcnt),
//   with the wait one iteration behind the issue so copies overlap WMMAs.

typedef __attribute__((ext_vector_type(2))) float v2f;
typedef __attribute__((ext_vector_type(8))) float v8f;
typedef __attribute__((vector_size(16))) int v4i_g;  // matches builtin param

__device__ __forceinline__ v8f wmma_f32(v2f a, v2f b, v8f c) {
  // (neg_a, A, neg_b, B, c_mod, C, reuse_a, reuse_b)
  return __builtin_amdgcn_wmma_f32_16x16x4_f32(false, a, false, b, (short)0, c,
                                               false, false);
}

#define GLOBAL_AS __attribute__((address_space(1)))
#define LDS_AS __attribute__((address_space(3)))

// generic->AS1 is numerically identity; generic->AS3 is low-32-bit LDS offset
// (flat LDS aperture: LDS address is the low 32 bits, per ISA 10.2).
__device__ __forceinline__ void async_copy_b128(const float* g, float* l) {
  __builtin_amdgcn_global_load_async_to_lds_b128(
      (GLOBAL_AS v4i_g*)(uintptr_t)g,
      (LDS_AS v4i_g*)(uint32_t)(uintptr_t)l, 0, 0);
}
__device__ __forceinline__ void async_copy_b32(const float* g, float* l) {
  __builtin_amdgcn_global_load_async_to_lds_b32(
      (GLOBAL_AS int*)(uintptr_t)g,
      (LDS_AS int*)(uint32_t)(uintptr_t)l, 0, 0);
}
__device__ __forceinline__ void wait_async0() {
  asm volatile("s_wait_asynccnt 0x0" ::: "memory");
}

// ---------------------------------------------------------------- LayerNorm
__global__ __launch_bounds__(256) void ln_kernel(const float* __restrict__ x,
                                                 const float* __restrict__ w,
                                                 const float* __restrict__ b,
                                                 float* __restrict__ y,
                                                 int Cdim) {
  __shared__ float wsum[8], wsum2[8];
  int row = blockIdx.x;
  const float* xr = x + (size_t)row * Cdim;
  float s = 0.f, s2 = 0.f;
  for (int i = threadIdx.x; i < Cdim; i += 256) {
    float v = xr[i];
    s += v;
    s2 += v * v;
  }
  for (int off = 16; off; off >>= 1) {
    s += __shfl_xor(s, off, 32);
    s2 += __shfl_xor(s2, off, 32);
  }
  int wid = threadIdx.x >> 5;
  if ((threadIdx.x & 31) == 0) {
    wsum[wid] = s;
    wsum2[wid] = s2;
  }
  __syncthreads();
  float ts = 0.f, ts2 = 0.f;
#pragma unroll
  for (int i = 0; i < 8; i++) {
    ts += wsum[i];
    ts2 += wsum2[i];
  }
  float mu = ts / (float)Cdim;
  float var = ts2 / (float)Cdim - mu * mu;
  float inv = rsqrtf(var + 1e-5f);
  for (int i = threadIdx.x; i < Cdim; i += 256) {
    float v = (xr[i] - mu) * inv;
    y[(size_t)row * Cdim + i] = v * w[i] + b[i];
  }
}

// ------------------------------------------------------------- WMMA GEMM
// C[M x N] = A[M x K] * B[K x N] (+bias)(+PReLU)(+resid), row major.
// Block: 256 threads (8 waves), tile 128x64; wave tile 32x32 (2x2 16x16).
#define GM_BM 128
#define GM_BN 64
#define GM_BK 16
#define GM_LDA 20  // 16 + 4 pad: b128 async stores stay 16B aligned
#define GM_LDB 18  // transposed B tile [n][k]; even pad keeps b64 reads aligned

__global__ __launch_bounds__(256) void gemm_wmma_kernel(
    const float* __restrict__ A, int lda, const float* __restrict__ B, int ldb,
    float* __restrict__ Cm, int ldc, int K, const float* __restrict__ bias,
    const float* __restrict__ resid, const float* __restrict__ prelu_a,
    long strideB, long strideC) {
  __shared__ float As[2][GM_BM * GM_LDA];
  __shared__ float Bs[2][GM_BN * GM_LDB];

  const float* Bz = B + (size_t)blockIdx.z * (size_t)strideB;
  float* Cz = Cm + (size_t)blockIdx.z * (size_t)strideC;

  const int tid = threadIdx.x;
  const int lane = tid & 31, wid = tid >> 5;
  const int waveM = wid >> 1, waveN = wid & 1;
  const int lh = lane & 15, hi = lane >> 4;
  const int rowBase = blockIdx.x * GM_BM;
  const int colBase = blockIdx.y * GM_BN;

  // per-thread async-copy addressing (running global pointers)
  const int ar = tid >> 2;       // A tile row 0..63 (and +64)
  const int ac = (tid & 3) * 4;  // A tile col group
  const float* ga0 = A + (size_t)(rowBase + ar) * lda + ac;
  const float* ga1 = ga0 + (size_t)64 * lda;
  const int bn = tid & 63;  // B tile col
  const int bk = tid >> 6;  // B tile row 0..3 (stride 4)
  const size_t bStep = (size_t)GM_BK * ldb;
  const float* gb0 = Bz + (size_t)bk * ldb + colBase + bn;
  const float* gb1 = gb0 + (size_t)4 * ldb;
  const float* gb2 = gb0 + (size_t)8 * ldb;
  const float* gb3 = gb0 + (size_t)12 * ldb;

  float* la0[2];
  float* la1[2];
  float* lb[2];
#pragma unroll
  for (int buf = 0; buf < 2; buf++) {
    la0[buf] = &As[buf][ar * GM_LDA + ac];
    la1[buf] = la0[buf] + 64 * GM_LDA;
    lb[buf] = &Bs[buf][bn * GM_LDB + bk];
  }

  auto issueTile = [&](int buf) {
    async_copy_b128(ga0, la0[buf]);
    async_copy_b128(ga1, la1[buf]);
    async_copy_b32(gb0, lb[buf]);
    async_copy_b32(gb1, lb[buf] + 4);
    async_copy_b32(gb2, lb[buf] + 8);
    async_copy_b32(gb3, lb[buf] + 12);
    ga0 += GM_BK;
    ga1 += GM_BK;
    gb0 += bStep;
    gb1 += bStep;
    gb2 += bStep;
    gb3 += bStep;
  };

  v8f acc[2][2] = {};
  const int nIter = K / GM_BK;
  issueTile(0);

  for (int it = 0; it < nIter; ++it) {
    const int buf = it & 1;
    wait_async0();    // my copies into As[buf]/Bs[buf] have landed
    __syncthreads();  // everyone's landed; buf^1 free to overwrite
    if (it + 1 < nIter) issueTile(buf ^ 1);  // overlaps the WMMAs below

    const float* As_ = &As[buf][0];
    const float* Bs_ = &Bs[buf][0];
#pragma unroll
    for (int kk = 0; kk < 4; kk++) {
      const int ks = kk * 4 + 2 * hi;
      v2f af[2], bf[2];
#pragma unroll
      for (int mi = 0; mi < 2; mi++)
        af[mi] = *(const v2f*)&As_[(waveM * 32 + mi * 16 + lh) * GM_LDA + ks];
#pragma unroll
      for (int ni = 0; ni < 2; ni++)
        bf[ni] = *(const v2f*)&Bs_[(waveN * 32 + ni * 16 + lh) * GM_LDB + ks];
#pragma unroll
      for (int mi = 0; mi < 2; mi++)
#pragma unroll
        for (int ni = 0; ni < 2; ni++)
          acc[mi][ni] = wmma_f32(af[mi], bf[ni], acc[mi][ni]);
    }
  }

  const float pa = prelu_a ? prelu_a[0] : 0.f;
#pragma unroll
  for (int mi = 0; mi < 2; mi++) {
#pragma unroll
    for (int ni = 0; ni < 2; ni++) {
      const int gcol = colBase + waveN * 32 + ni * 16 + lh;
      const float bv = bias ? bias[gcol] : 0.f;
#pragma unroll
      for (int r = 0; r < 8; r++) {
        const int grow = rowBase + waveM * 32 + mi * 16 + r + 8 * hi;
        float v = acc[mi][ni][r] + bv;
        if (prelu_a) v = (v > 0.f) ? v : pa * v;
        if (resid) v += resid[(size_t)grow * ldc + gcol];
        Cz[(size_t)grow * ldc + gcol] = v;
      }
    }
  }
}

// ----------------------------------------------- flash attention (causal)
// wei = softmax(mask(K Q^T / 32)); O = wei @ V.  One wave per 16 t-rows per
// (b,h). K/Q/V stored [B*T, 1024] with head h at column h*64.
__global__ __launch_bounds__(128) void attn_kernel(const float* __restrict__ Kb,
                                                   const float* __restrict__ Qb,
                                                   const float* __restrict__ Vb,
                                                   float* __restrict__ Ob) {
  __shared__ float Ps[4][16 * 18];  // per-wave P transpose staging
  const int lane = threadIdx.x & 31, wid = threadIdx.x >> 5;
  const int w = blockIdx.x * 4 + wid;
  const int t0 = (w & 63) * 16;
  const int bh = w >> 6;
  const int bb = bh >> 4, hh = bh & 15;
  const int lh = lane & 15, hi = lane >> 4;
  const int ld = 1024;

  const size_t base = (size_t)bb * 1024 * ld + hh * 64;
  const float* Kp = Kb + base;
  const float* Qp = Qb + base;
  const float* Vp = Vb + base;

  // K-tile (16 x 64) held as 16 A-fragments
  v2f ak[16];
#pragma unroll
  for (int kk = 0; kk < 16; kk++)
    ak[kk] = *(const v2f*)&Kp[(size_t)(t0 + lh) * ld + kk * 4 + 2 * hi];

  float rmax[8], rsum[8];
#pragma unroll
  for (int r = 0; r < 8; r++) {
    rmax[r] = -1e30f;
    rsum[r] = 0.f;
  }
  v8f o[4] = {};
  float* myPs = Ps[wid];

  for (int s0 = 0; s0 <= t0; s0 += 16) {
    v8f s = {};
#pragma unroll
    for (int kk = 0; kk < 16; kk++) {
      v2f bq = *(const v2f*)&Qp[(size_t)(s0 + lh) * ld + kk * 4 + 2 * hi];
      s = wmma_f32(ak[kk], bq, s);
    }
    const bool diag = (s0 == t0);
#pragma unroll
    for (int r = 0; r < 8; r++) {
      float v = s[r] * 0.03125f;  // C^-0.5 = 1/32
      if (diag && (lh > r + 8 * hi)) v = -1e30f;  // causal: s<=t
      s[r] = v;
    }
#pragma unroll
    for (int r = 0; r < 8; r++) {
      float v = s[r];
      for (int off = 1; off < 16; off <<= 1) v = fmaxf(v, __shfl_xor(v, off, 32));
      const float mnew = fmaxf(rmax[r], v);
      const float corr = __expf(rmax[r] - mnew);
      const float p = __expf(s[r] - mnew);
      float psum = p;
      for (int off = 1; off < 16; off <<= 1) psum += __shfl_xor(psum, off, 32);
      rsum[r] = rsum[r] * corr + psum;
      rmax[r] = mnew;
      s[r] = p;
#pragma unroll
      for (int dt = 0; dt < 4; dt++) o[dt][r] *= corr;
    }
    // P: C-layout -> A-layout via per-wave LDS tile (DS in-order per wave)
#pragma unroll
    for (int r = 0; r < 8; r++) myPs[(r + 8 * hi) * 18 + lh] = s[r];
#pragma unroll
    for (int kk = 0; kk < 4; kk++) {
      const int ks = kk * 4 + 2 * hi;
      v2f pa = *(const v2f*)&myPs[lh * 18 + ks];
#pragma unroll
      for (int dt = 0; dt < 4; dt++) {
        v2f bv;
        bv.x = Vp[(size_t)(s0 + ks) * ld + dt * 16 + lh];
        bv.y = Vp[(size_t)(s0 + ks + 1) * ld + dt * 16 + lh];
        o[dt] = wmma_f32(pa, bv, o[dt]);
      }
    }
  }
#pragma unroll
  for (int r = 0; r < 8; r++) {
    const float inv = 1.f / rsum[r];
#pragma unroll
    for (int dt = 0; dt < 4; dt++) o[dt][r] *= inv;
  }
#pragma unroll
  for (int dt = 0; dt < 4; dt++)
#pragma unroll
    for (int r = 0; r < 8; r++)
      Ob[((size_t)bb * 1024 + t0 + r + 8 * hi) * 1024 + hh * 64 + dt * 16 + lh] =
          o[dt][r];
}

// ------------------------------------------------------------------ launch
extern "C" void kernel_launch(void* const* d_in, const int* in_sizes, int n_in,
                              void* d_out, int out_size, void* d_ws,
                              size_t ws_size, hipStream_t stream) {
  (void)in_sizes; (void)n_in; (void)out_size; (void)ws_size;
  const float* x = (const float*)d_in[0];
  const float* ln1_w = (const float*)d_in[1];
  const float* ln1_b = (const float*)d_in[2];
  const float* Wk = (const float*)d_in[3];
  const float* Wq = (const float*)d_in[4];
  const float* Wv = (const float*)d_in[5];
  const float* Wo = (const float*)d_in[6];
  const float* bo = (const float*)d_in[7];
  const float* ln2_w = (const float*)d_in[8];
  const float* ln2_b = (const float*)d_in[9];
  const float* W1 = (const float*)d_in[10];
  const float* b1 = (const float*)d_in[11];
  const float* prelu_a = (const float*)d_in[12];
  const float* W2 = (const float*)d_in[13];
  const float* b2 = (const float*)d_in[14];
  float* out = (float*)d_out;

  const int Bn = 4, T = 1024, C = 1024, H = 16, HS = 64;
  const int M = Bn * T;  // 4096

  float* ws = (float*)d_ws;
  size_t off = 0;
  float* hbuf = ws + off; off += (size_t)M * C;   // h / h2 (reused)
  float* qbuf = ws + off; off += (size_t)M * C;
  float* kbuf = ws + off; off += (size_t)M * C;
  float* vbuf = ws + off; off += (size_t)M * C;
  float* obuf = ws + off; off += (size_t)M * C;
  float* x1   = ws + off; off += (size_t)M * C;
  float* fbuf = ws + off; off += (size_t)M * 4 * C;

  // 1) h = LN1(x)
  ln_kernel<<<M, 256, 0, stream>>>(x, ln1_w, ln1_b, hbuf, C);

  // 2) per-head QKV projections: [4096x1024] x [1024x64] per head (grid.z)
  dim3 gq(M / GM_BM, 1, H);
  gemm_wmma_kernel<<<gq, 256, 0, stream>>>(hbuf, C, Wq, HS, qbuf, C, C,
                                           nullptr, nullptr, nullptr,
                                           (long)C * HS, (long)HS);
  gemm_wmma_kernel<<<gq, 256, 0, stream>>>(hbuf, C, Wk, HS, kbuf, C, C,
                                           nullptr, nullptr, nullptr,
                                           (long)C * HS, (long)HS);
  gemm_wmma_kernel<<<gq, 256, 0, stream>>>(hbuf, C, Wv, HS, vbuf, C, C,
                                           nullptr, nullptr, nullptr,
                                           (long)C * HS, (long)HS);

  // 3) causal attention (flash-style, WMMA)
  attn_kernel<<<(Bn * H * (T / 16)) / 4, 128, 0, stream>>>(kbuf, qbuf, vbuf,
                                                           obuf);

  // 4) x1 = x + O @ Wo + bo
  dim3 gp(M / GM_BM, C / GM_BN, 1);
  gemm_wmma_kernel<<<gp, 256, 0, stream>>>(obuf, C, Wo, C, x1, C, C, bo, x,
                                           nullptr, 0, 0);

  // 5) h2 = LN2(x1)
  ln_kernel<<<M, 256, 0, stream>>>(x1, ln2_w, ln2_b, hbuf, C);

  // 6) f = PReLU(h2 @ W1 + b1)
  dim3 g1(M / GM_BM, (4 * C) / GM_BN, 1);
  gemm_wmma_kernel<<<g1, 256, 0, stream>>>(hbuf, C, W1, 4 * C, fbuf, 4 * C, C,
                                           b1, nullptr, prelu_a, 0, 0);

  // 7) out = x1 + f @ W2 + b2
  gemm_wmma_kernel<<<gp, 256, 0, stream>>>(fbuf, 4 * C, W2, C, out, C, 4 * C,
                                           b2, x1, nullptr, 0, 0);
}